// MutationEncoder_26731876450407
// MI455X (gfx1250) — compile-verified
//
#include <hip/hip_runtime.h>

typedef float v2f __attribute__((ext_vector_type(2)));
typedef float v8f __attribute__((ext_vector_type(8)));

#define N_AA 22
#define NPOS 29
#define DIM  32

// 0-indexed positions: 13 major then 16 accessory, padded to 32
__constant__ int c_pos[32] = {
    29, 31, 32, 45, 46, 47, 49, 53, 75, 81, 83, 87, 89,
     9, 10, 15, 19, 23, 34, 35, 52, 61, 62, 70, 72, 73, 76, 84, 92,
     0,  0,  0};

// Sum of 21 contiguous floats starting at xr + pos*22 (8-byte aligned: row
// stride 2178 floats = 8712 B and 22*4 B are both multiples of 8).
__device__ __forceinline__ float chunk_sum(const float* __restrict__ xr, int pos) {
    const v2f* p2 = (const v2f*)(xr + pos * N_AA);
    float s = 0.f;
#pragma unroll
    for (int j = 0; j < 10; ++j) { v2f v = p2[j]; s += v.x + v.y; }
    s += ((const float*)p2)[20];
    return s;
}

// Stage 1: fold embeddings into W_combine.
// ptab[k][n] = sum_d E_k[d] * W[n][d or 32+d]   (k<13 major, 13<=k<29 acc, else 0)
__global__ __launch_bounds__(1024) void build_table_kernel(
    const float* __restrict__ me, const float* __restrict__ ae,
    const float* __restrict__ W,  const float* __restrict__ b,
    float* __restrict__ ptab, float* __restrict__ bias) {
    const int t = threadIdx.x;     // 0..1023
    const int k = t >> 5;
    const int n = t & 31;
    float s = 0.f;
    if (k < 13) {
#pragma unroll
        for (int d = 0; d < DIM; ++d) s += me[k * DIM + d] * W[n * 64 + d];
    } else if (k < NPOS) {
        const int j = k - 13;
#pragma unroll
        for (int d = 0; d < DIM; ++d) s += ae[j * DIM + d] * W[n * 64 + DIM + d];
    }
    ptab[t] = s;
    if (t < DIM) bias[t] = b[t];
}

// Stage 2: one wave per 16-row tile. Each lane computes exactly the mask
// values its WMMA A-fragment needs (no LDS, no shuffles), then 16x
// v_wmma_f32_16x16x4_f32 accumulate out = mut @ ptab, + bias, store.
__global__ __launch_bounds__(256) void mut_gemm_wmma_kernel(
    const float* __restrict__ x, const float* __restrict__ ptab,
    const float* __restrict__ bias, float* __restrict__ out, int nrows) {
    const int lane = threadIdx.x & 31;
    const int wave = threadIdx.x >> 5;
    const int tile = blockIdx.x * 8 + wave;
    const int base = tile * 16;
    if (base >= nrows) return;            // wave-uniform: EXEC stays all-ones

    const int half = lane >> 4;           // 0: K=4t,4t+1   1: K=4t+2,4t+3
    const int l16  = lane & 15;           // row within tile (A), col within half (B/D)
    int row = base + l16;
    if (row >= nrows) row = nrows - 1;    // safe clamp for ragged tail reads
    const float* xr = x + (size_t)row * (99 * N_AA);

    // ---- A fragments: mut values in WMMA 16x4 fp32 layout ----
    v2f a[8];
#pragma unroll
    for (int t = 0; t < 8; ++t) {
        const int k0 = 4 * t + 2 * half;
        float m0 = 0.f, m1 = 0.f;
        if (k0 < NPOS)     m0 = chunk_sum(xr, c_pos[k0])     > 0.f ? 1.f : 0.f;
        if (k0 + 1 < NPOS) m1 = chunk_sum(xr, c_pos[k0 + 1]) > 0.f ? 1.f : 0.f;
        a[t].x = m0;
        a[t].y = m1;
    }

    // ---- GEMM: D(16x32) = A(16x32) x Ptab(32x32), two N-halves ----
    v8f d0 = {};
    v8f d1 = {};
#pragma unroll
    for (int t = 0; t < 8; ++t) {
        const int k0 = 4 * t + 2 * half;
        v2f b0, b1;
        b0.x = ptab[k0 * DIM + l16];
        b0.y = ptab[(k0 + 1) * DIM + l16];
        b1.x = ptab[k0 * DIM + 16 + l16];
        b1.y = ptab[(k0 + 1) * DIM + 16 + l16];
        d0 = __builtin_amdgcn_wmma_f32_16x16x4_f32(false, a[t], false, b0,
                                                   (short)0, d0, false, false);
        d1 = __builtin_amdgcn_wmma_f32_16x16x4_f32(false, a[t], false, b1,
                                                   (short)0, d1, false, false);
    }

    // ---- bias + store: VGPR j holds rows j / j+8, lane -> column ----
    const float bn0 = bias[l16];
    const float bn1 = bias[16 + l16];
    float* obase = out + (size_t)base * DIM;
#pragma unroll
    for (int j = 0; j < 8; ++j) {
        const int m = j + half * 8;
        if (base + m < nrows) {
            obase[m * DIM + l16]      = d0[j] + bn0;
            obase[m * DIM + 16 + l16] = d1[j] + bn1;
        }
    }
}

extern "C" void kernel_launch(void* const* d_in, const int* in_sizes, int n_in,
                              void* d_out, int out_size, void* d_ws, size_t ws_size,
                              hipStream_t stream) {
    const float* x  = (const float*)d_in[0];   // (B, 99*22) f32
    const float* me = (const float*)d_in[1];   // (13, 32)   f32
    const float* ae = (const float*)d_in[2];   // (16, 32)   f32
    const float* W  = (const float*)d_in[3];   // (32, 64)   f32
    const float* b  = (const float*)d_in[4];   // (32,)      f32
    float* out  = (float*)d_out;               // (B, 32)    f32
    float* ptab = (float*)d_ws;                // 32x32 folded table (4 KB)
    float* bias = ptab + 1024;                 // 32 floats

    const int nrows = in_sizes[0] / (99 * N_AA);

    build_table_kernel<<<1, 1024, 0, stream>>>(me, ae, W, b, ptab, bias);

    const int tiles  = (nrows + 15) / 16;
    const int blocks = (tiles + 7) / 8;
    mut_gemm_wmma_kernel<<<blocks, 256, 0, stream>>>(x, ptab, bias, out, nrows);
}